// NTMReadController_41609643164442
// MI455X (gfx1250) — compile-verified
//
#include <hip/hip_runtime.h>
#include <math.h>

// Problem constants (from reference): B=64, E=512, N=8192, M=64, C=M+6=70
#define B_  64
#define E_  512
#define N_  8192
#define M_  64
#define C_  70
#define EPS_ 1e-8f

typedef __attribute__((ext_vector_type(2))) float v2f;
typedef __attribute__((ext_vector_type(8))) float v8f;

// ---------------------------------------------------------------------------
// Kernel 1: addr = emb @ W + b  via V_WMMA_F32_16X16X4_F32
// Output [64,70] tiled as 4x5 16x16 tiles, one wave (32 threads) per tile.
// A-frag layout (16x4 f32): lanes 0-15 -> K={k0,k0+1}, lanes 16-31 -> K={k0+2,k0+3}
// B-frag layout (4x16 f32): symmetric, col = lane%16
// C/D layout: VGPR j -> row (j + 8*(lane>=16)), col = lane%16
// Column guard is hoisted out of the K-loop: clamp index (stays in-bounds) and
// multiply by a {0,1} lane mask -> branch-free inner loop, EXEC stays all-1s.
// ---------------------------------------------------------------------------
__global__ __launch_bounds__(32)
void ntm_addr_gemm_wmma(const float* __restrict__ emb,
                        const float* __restrict__ W,
                        const float* __restrict__ bias,
                        float* __restrict__ addr)
{
    const int tile = blockIdx.x;       // 0..19
    const int tm   = tile / 5;         // row tile 0..3
    const int tn   = tile % 5;         // col tile 0..4
    const int lane = threadIdx.x;      // 0..31
    const int half = lane >> 4;        // 0|1
    const int l16  = lane & 15;
    const int row  = tm * 16 + l16;    // A row for this lane
    const int col  = tn * 16 + l16;    // B/C column for this lane
    const bool colOK = (col < C_);
    const int  colC  = colOK ? col : (C_ - 1);   // clamped, always in-bounds
    const float cmask = colOK ? 1.0f : 0.0f;     // zero out padded columns

    const float* __restrict__ arow = emb + row * E_;
    const float* __restrict__ bcol = W + colC;

    v8f acc = {};
    for (int k0 = 0; k0 < E_; k0 += 4) {
        const int ka = k0 + half * 2;
        v2f a;
        a.x = arow[ka];
        a.y = arow[ka + 1];
        v2f bm;
        bm.x = bcol[ka * C_] * cmask;
        bm.y = bcol[(ka + 1) * C_] * cmask;
        acc = __builtin_amdgcn_wmma_f32_16x16x4_f32(
            /*neg_a=*/false, a, /*neg_b=*/false, bm,
            /*c_mod=*/(short)0, acc, /*reuse_a=*/false, /*reuse_b=*/false);
    }
    if (colOK) {
        const float bv = bias[col];
#pragma unroll
        for (int j = 0; j < 8; ++j) {
            const int r = tm * 16 + j + half * 8;
            addr[r * C_ + col] = acc[j] + bv;
        }
    }
}

// ---------------------------------------------------------------------------
// Kernel 2: per-batch activations.
// scal[b][0..7] = beta, g, s0, s1, s2, y, k_norm, (pad)
// ---------------------------------------------------------------------------
__device__ __forceinline__ float softplus_f(float x) {
    // numerically stable log1p(exp(x))
    return fmaxf(x, 0.0f) + log1pf(expf(-fabsf(x)));
}

__global__ __launch_bounds__(64)
void ntm_prep_scalars(const float* __restrict__ addr, float* __restrict__ scal)
{
    const int b = blockIdx.x * blockDim.x + threadIdx.x;
    if (b >= B_) return;
    const float* a = addr + b * C_;
    const float beta = softplus_f(a[M_ + 0]);
    const float g    = 1.0f / (1.0f + expf(-a[M_ + 1]));
    const float s0i = a[M_ + 2], s1i = a[M_ + 3], s2i = a[M_ + 4];
    const float mx = fmaxf(s0i, fmaxf(s1i, s2i));
    const float e0 = expf(s0i - mx), e1 = expf(s1i - mx), e2 = expf(s2i - mx);
    const float inv = 1.0f / (e0 + e1 + e2);
    const float y = 1.0f + softplus_f(a[M_ + 5]);
    float ks = 0.0f;
#pragma unroll
    for (int m = 0; m < M_; ++m) { const float v = a[m]; ks += v * v; }
    float* s = scal + b * 8;
    s[0] = beta; s[1] = g; s[2] = e0 * inv; s[3] = e1 * inv; s[4] = e2 * inv;
    s[5] = y;    s[6] = sqrtf(ks); s[7] = 0.0f;
}

// ---------------------------------------------------------------------------
// Kernel 3: full NTM addressing pipeline. One block per batch, 1024 threads.
// LDS: Ash[8192] holds (in sequence) logits -> exp -> w_g -> w_pow -> w.
// ---------------------------------------------------------------------------
#define T3 1024
#define RPT (N_ / T3)   // rows per thread = 8

__global__ __launch_bounds__(T3)
void ntm_address_read(const float* __restrict__ addr,
                      const float* __restrict__ scal,
                      const float* __restrict__ w_prev,
                      const float* __restrict__ mem,
                      float* __restrict__ out_md,   // [B,M]
                      float* __restrict__ out_w)    // [B,N]
{
    __shared__ float Ash[N_];
    __shared__ float kv[M_];
    __shared__ float red[T3];

    const int b   = blockIdx.x;
    const int tid = threadIdx.x;

    // broadcast per-batch scalars + stage k in LDS
    const float* sc = scal + b * 8;
    const float beta = sc[0], g = sc[1], s0 = sc[2], s1 = sc[3], s2 = sc[4];
    const float yexp = sc[5], knorm = sc[6];
    if (tid < M_) kv[tid] = addr[b * C_ + tid];
    __syncthreads();

    const float* __restrict__ memB = mem + (size_t)b * N_ * M_;

    // ---- pass 1: fused dot + row-norm -> scaled cosine logits ----
    float lmax = -3.4e38f;
#pragma unroll
    for (int it = 0; it < RPT; ++it) {
        const int n = tid + it * T3;
        const float4* rp = reinterpret_cast<const float4*>(memB + (size_t)n * M_);
        const float4* kp = reinterpret_cast<const float4*>(kv);
        float dot = 0.0f, ss = 0.0f;
#pragma unroll
        for (int j = 0; j < M_ / 4; ++j) {
            const float4 v = rp[j];
            const float4 kk = kp[j];
            dot += v.x * kk.x + v.y * kk.y + v.z * kk.z + v.w * kk.w;
            ss  += v.x * v.x + v.y * v.y + v.z * v.z + v.w * v.w;
        }
        const float sim = dot / (sqrtf(ss) * knorm + EPS_);
        const float t = beta * sim;
        Ash[n] = t;
        lmax = fmaxf(lmax, t);
    }

    // ---- block max reduction ----
    red[tid] = lmax; __syncthreads();
    for (int s = T3 / 2; s > 0; s >>= 1) {
        if (tid < s) red[tid] = fmaxf(red[tid], red[tid + s]);
        __syncthreads();
    }
    const float mx = red[0]; __syncthreads();

    // ---- exp + block sum ----
    float lsum = 0.0f;
#pragma unroll
    for (int it = 0; it < RPT; ++it) {
        const int n = tid + it * T3;
        const float e = expf(Ash[n] - mx);
        Ash[n] = e;
        lsum += e;
    }
    red[tid] = lsum; __syncthreads();
    for (int s = T3 / 2; s > 0; s >>= 1) {
        if (tid < s) red[tid] += red[tid + s];
        __syncthreads();
    }
    const float invS = 1.0f / red[0]; __syncthreads();

    // ---- interpolation: w_g = g*w_c + (1-g)*w_prev (in place) ----
    const float* __restrict__ wpB = w_prev + (size_t)b * N_;
#pragma unroll
    for (int it = 0; it < RPT; ++it) {
        const int n = tid + it * T3;
        Ash[n] = g * (Ash[n] * invS) + (1.0f - g) * wpB[n];
    }
    __syncthreads();

    // ---- circular 3-tap conv + sharpening (read phase -> barrier -> write) ----
    float wp[RPT];
#pragma unroll
    for (int it = 0; it < RPT; ++it) {
        const int n = tid + it * T3;
        const float lft = Ash[(n + N_ - 1) & (N_ - 1)];
        const float mid = Ash[n];
        const float rgt = Ash[(n + 1) & (N_ - 1)];
        const float wsv = s0 * lft + s1 * mid + s2 * rgt;
        wp[it] = powf(wsv, yexp);
    }
    __syncthreads();
    float psum = 0.0f;
#pragma unroll
    for (int it = 0; it < RPT; ++it) {
        const int n = tid + it * T3;
        Ash[n] = wp[it];
        psum += wp[it];
    }
    red[tid] = psum; __syncthreads();
    for (int s = T3 / 2; s > 0; s >>= 1) {
        if (tid < s) red[tid] += red[tid + s];
        __syncthreads();
    }
    const float invP = 1.0f / (red[0] + EPS_); __syncthreads();

    // ---- normalize -> final w (keep in LDS, emit to d_out) ----
    float* __restrict__ owB = out_w + (size_t)b * N_;
#pragma unroll
    for (int it = 0; it < RPT; ++it) {
        const int n = tid + it * T3;
        const float wv = Ash[n] * invP;
        Ash[n] = wv;
        owB[n] = wv;
    }
    __syncthreads();

    // ---- pass 2: memory_data[b][m] = sum_n w[n] * mem[b][n][m] ----
    // 16 groups of 64 threads; lanes map to consecutive m (coalesced).
    const int grp = tid >> 6;          // 0..15
    const int m   = tid & 63;          // 0..63
    const int n0  = grp * (N_ / 16);   // 512-row chunk
    float acc = 0.0f;
#pragma unroll 8
    for (int n = n0; n < n0 + N_ / 16; ++n) {
        acc += Ash[n] * memB[(size_t)n * M_ + m];
    }
    red[tid] = acc; __syncthreads();
    if (tid < M_) {
        float tot = 0.0f;
#pragma unroll
        for (int gg = 0; gg < 16; ++gg) tot += red[gg * 64 + tid];
        out_md[b * M_ + tid] = tot;
    }
}

// ---------------------------------------------------------------------------
// Host-side launcher
// ---------------------------------------------------------------------------
extern "C" void kernel_launch(void* const* d_in, const int* in_sizes, int n_in,
                              void* d_out, int out_size, void* d_ws, size_t ws_size,
                              hipStream_t stream)
{
    const float* emb    = (const float*)d_in[0];  // [64,512]
    const float* w_prev = (const float*)d_in[1];  // [64,8192]
    const float* mem    = (const float*)d_in[2];  // [64,8192,64]
    const float* W      = (const float*)d_in[3];  // [512,70]
    const float* bias   = (const float*)d_in[4];  // [70]

    float* out    = (float*)d_out;
    float* out_md = out;                // [64,64]   (first return value)
    float* out_w  = out + B_ * M_;      // [64,8192] (second return value)

    float* ws   = (float*)d_ws;
    float* addr = ws;                   // [64,70]
    float* scal = ws + B_ * C_;         // [64,8]

    ntm_addr_gemm_wmma<<<20, 32, 0, stream>>>(emb, W, bias, addr);
    ntm_prep_scalars<<<(B_ + 63) / 64, 64, 0, stream>>>(addr, scal);
    ntm_address_read<<<B_, T3, 0, stream>>>(addr, scal, w_prev, mem, out_md, out_w);
}